// DenseMoE_85873576116904
// MI455X (gfx1250) — compile-verified
//
#include <hip/hip_runtime.h>
#include <hip/hip_bf16.h>
#include <stdint.h>
#include <stddef.h>

typedef __attribute__((ext_vector_type(2))) float v2f;
typedef __attribute__((ext_vector_type(8))) float v8f;
typedef int v4i_t __attribute__((vector_size(16)));
typedef __attribute__((address_space(1))) v4i_t* gptr_v4i;
typedef __attribute__((address_space(3))) v4i_t* lptr_v4i;

#define NBATCH 8
#define CIN    64
#define HH     128
#define WW     128
#define NEXP   8
#define COUT   64
#define MTOT   (NEXP * COUT)   /* 512 flattened output channels */
#define HW     (HH * WW)       /* 16384 pixels per image        */
#define MT     128             /* M tile per block              */
#define ASTR   68              /* A LDS row stride (floats)     */
#define BSTR   136             /* B LDS row stride (floats)     */

#if __has_builtin(__builtin_amdgcn_global_load_async_to_lds_b128) && \
    __has_builtin(__builtin_amdgcn_s_wait_asynccnt)
#define HAVE_ASYNC_LDS 1
#else
#define HAVE_ASYNC_LDS 0
#endif

// ---------------------------------------------------------------------------
// Kernel 1: gating. 1x1 conv (Cin=64 -> K=8), softmax over experts, top-2
// mask. One thread per pixel; gate weights staged in LDS (2 KB).
// ---------------------------------------------------------------------------
__global__ __launch_bounds__(256) void moe_gate_kernel(
    const float* __restrict__ x, const float* __restrict__ gate_w,
    const float* __restrict__ gate_b, float* __restrict__ out_gate,
    float* __restrict__ out_mask)
{
    __shared__ float gw[NEXP * CIN];
    const int tid = threadIdx.x;
    for (int i = tid; i < NEXP * CIN; i += 256) gw[i] = gate_w[i];
    __syncthreads();

    const int pid = blockIdx.x * 256 + tid;        // 0 .. 131071
    const int b   = pid >> 14;
    const int hw  = pid & (HW - 1);

    float lg[NEXP];
#pragma unroll
    for (int e = 0; e < NEXP; e++) lg[e] = gate_b[e];

    const float* xp = x + (size_t)b * CIN * HW + hw;
    for (int c = 0; c < CIN; c++) {
        const float xv = xp[(size_t)c * HW];
#pragma unroll
        for (int e = 0; e < NEXP; e++) lg[e] = fmaf(xv, gw[e * CIN + c], lg[e]);
    }

    // softmax over 8 experts
    float mx = lg[0];
#pragma unroll
    for (int e = 1; e < NEXP; e++) mx = fmaxf(mx, lg[e]);
    float p[NEXP], s = 0.0f;
#pragma unroll
    for (int e = 0; e < NEXP; e++) { p[e] = __expf(lg[e] - mx); s += p[e]; }
    const float inv = 1.0f / s;
#pragma unroll
    for (int e = 0; e < NEXP; e++) {
        p[e] *= inv;
        out_gate[((size_t)b * NEXP + e) * HW + hw] = p[e];
    }

    // top-2 (first occurrence wins ties, matching lax.top_k)
    int i0 = 0; float v0 = p[0];
#pragma unroll
    for (int e = 1; e < NEXP; e++) if (p[e] > v0) { v0 = p[e]; i0 = e; }
    int i1 = (i0 == 0) ? 1 : 0; float v1 = p[i1];
#pragma unroll
    for (int e = 0; e < NEXP; e++)
        if (e != i0 && p[e] > v1) { v1 = p[e]; i1 = e; }

#pragma unroll
    for (int e = 0; e < NEXP; e++)
        out_mask[((size_t)b * NEXP + e) * HW + hw] =
            (e == i0 || e == i1) ? 1.0f : 0.0f;
}

// ---------------------------------------------------------------------------
// Kernel 2: dense-MoE 3x3 conv as implicit GEMM with V_WMMA_F32_16X16X4_F32.
// Block = 256 threads (8 waves). Tile: 128 out-channels x 128 pixels (one
// image row). Outer loop over the 9 (kh,kw) taps; per tap GEMM over Cin=64.
//   A (16x4  f32): lane(l16,half), vgpr v -> A[l16][2*half + v]
//   B (4x16  f32): lane(l16,half), vgpr v -> B[c0 + 2*half + v][n0 + l16]
//   D (16x16 f32): lane(l16,half), vgpr v -> D[v + 8*half][n0 + l16]
// ---------------------------------------------------------------------------
__global__ __launch_bounds__(256) void moe_expert_conv_wmma(
    const float* __restrict__ x, const float* __restrict__ ew,
    const float* __restrict__ eb, float* __restrict__ out)
{
    __shared__ float As[MT][ASTR];    // [out-channel][cin] for current tap
    __shared__ float Bs[CIN][BSTR];   // [cin][col]; col = wx + 4, wx in [-1,128]

    const int tid   = threadIdx.x;
    const int lane  = tid & 31;
    const int wv    = tid >> 5;            // wave id 0..7
    const int l16   = lane & 15;
    const int half  = lane >> 4;
    const int m0    = blockIdx.x * MT;     // 0,128,256,384
    const int h     = blockIdx.y;          // image row
    const int b     = blockIdx.z;          // batch
    const int waveM = (wv & 1) * 64;       // 2 waves tile M
    const int waveN = (wv >> 1) * 32;      // 4 waves tile N

    v8f acc[4][2];
#pragma unroll
    for (int i = 0; i < 4; i++)
#pragma unroll
        for (int j = 0; j < 2; j++)
            acc[i][j] = (v8f){0, 0, 0, 0, 0, 0, 0, 0};

    for (int kh = 0; kh < 3; kh++) {
        const int hy = h + kh - 1;
        __syncthreads();  // previous tap's reads of Bs are done

        // zero halo columns (wx = -1 at col 3, wx = 128 at col 132)
        if (tid < 128) {
            const int r = tid & 63;
            Bs[r][(tid >> 6) ? 132 : 3] = 0.0f;
        }

        if (hy >= 0 && hy < HH) {
#if HAVE_ASYNC_LDS
            // async global -> LDS, 16B (4-float) granules, tracked by ASYNCcnt
            for (int idx = tid; idx < CIN * 32; idx += 256) {
                const int cin = idx >> 5;
                const int w4  = (idx & 31) << 2;
                const float* src =
                    &x[(((size_t)b * CIN + cin) * HH + hy) * WW + w4];
                __builtin_amdgcn_global_load_async_to_lds_b128(
                    (gptr_v4i)src, (lptr_v4i)&Bs[cin][4 + w4], 0, 0);
            }
#else
            for (int idx = tid; idx < CIN * WW; idx += 256) {
                const int cin = idx >> 7;
                const int wx  = idx & 127;
                Bs[cin][4 + wx] =
                    x[(((size_t)b * CIN + cin) * HH + hy) * WW + wx];
            }
#endif
        } else {
            for (int idx = tid; idx < CIN * WW; idx += 256) {
                const int cin = idx >> 7;
                const int wx  = idx & 127;
                Bs[cin][4 + wx] = 0.0f;
            }
        }

        for (int kw = 0; kw < 3; kw++) {
            const int kk = kh * 3 + kw;
            __syncthreads();  // previous tap's reads of As are done

            // stage weights for this tap: As[m][cin] = W[m0+m][cin][kh][kw]
            for (int idx = tid; idx < MT * CIN; idx += 256) {
                const int mi  = idx >> 6;
                const int cin = idx & 63;
                As[mi][cin] = ew[((size_t)(m0 + mi) * CIN + cin) * 9 + kk];
            }
#if HAVE_ASYNC_LDS
            if (kw == 0) __builtin_amdgcn_s_wait_asynccnt(0);
#endif
            __syncthreads();  // publish As (and Bs when kw==0)

            // GEMM over Cin in K=4 steps: 16 steps x 8 tiles = 128 WMMA
            for (int c0 = 0; c0 < CIN; c0 += 4) {
                v2f a[4], bf[2];
#pragma unroll
                for (int i = 0; i < 4; i++)
                    a[i] = *(const v2f*)&As[waveM + 16 * i + l16]
                                          [c0 + 2 * half];
#pragma unroll
                for (int j = 0; j < 2; j++) {
                    const int col = waveN + 16 * j + l16 + kw + 3;
                    const int r   = c0 + 2 * half;
                    bf[j].x = Bs[r][col];
                    bf[j].y = Bs[r + 1][col];
                }
#pragma unroll
                for (int i = 0; i < 4; i++)
#pragma unroll
                    for (int j = 0; j < 2; j++)
                        acc[i][j] = __builtin_amdgcn_wmma_f32_16x16x4_f32(
                            false, a[i], false, bf[j],
                            (short)0, acc[i][j], false, false);
            }
        }
    }

    // epilogue: bias + store expert_out[b][m][h][w]
#pragma unroll
    for (int i = 0; i < 4; i++) {
        const int mbase = m0 + waveM + 16 * i + half * 8;
#pragma unroll
        for (int j = 0; j < 2; j++) {
            const int n = waveN + 16 * j + l16;
#pragma unroll
            for (int v = 0; v < 8; v++) {
                const int m = mbase + v;
                out[(((size_t)b * MTOT + m) * HH + h) * WW + n] =
                    acc[i][j][v] + eb[m];
            }
        }
    }
}

// ---------------------------------------------------------------------------
// Kernel 3: combine. Re-derive top-2 from gate_probs (already in d_out),
// weighted sum of the two selected experts' 64 channels -> prediction.
// ---------------------------------------------------------------------------
__global__ __launch_bounds__(256) void moe_combine_kernel(
    const float* __restrict__ gate, const float* __restrict__ eo,
    float* __restrict__ pred)
{
    const int pid = blockIdx.x * 256 + threadIdx.x;
    const int b   = pid >> 14;
    const int hw  = pid & (HW - 1);

    float p[NEXP];
#pragma unroll
    for (int e = 0; e < NEXP; e++)
        p[e] = gate[((size_t)b * NEXP + e) * HW + hw];

    int i0 = 0; float v0 = p[0];
#pragma unroll
    for (int e = 1; e < NEXP; e++) if (p[e] > v0) { v0 = p[e]; i0 = e; }
    int i1 = (i0 == 0) ? 1 : 0; float v1 = p[i1];
#pragma unroll
    for (int e = 0; e < NEXP; e++)
        if (e != i0 && p[e] > v1) { v1 = p[e]; i1 = e; }

    const float s  = v0 + v1 + 1e-8f;
    const float w0 = v0 / s, w1 = v1 / s;

    const float* e0 = eo + ((size_t)b * MTOT + i0 * COUT) * HW + hw;
    const float* e1 = eo + ((size_t)b * MTOT + i1 * COUT) * HW + hw;
    float* op = pred + (size_t)b * COUT * HW + hw;
    for (int c = 0; c < COUT; c++)
        op[(size_t)c * HW] =
            w0 * e0[(size_t)c * HW] + w1 * e1[(size_t)c * HW];
}

// ---------------------------------------------------------------------------
extern "C" void kernel_launch(void* const* d_in, const int* in_sizes, int n_in,
                              void* d_out, int out_size, void* d_ws,
                              size_t ws_size, hipStream_t stream)
{
    (void)in_sizes; (void)n_in; (void)d_ws; (void)ws_size; (void)out_size;

    const float* x        = (const float*)d_in[0];
    const float* gate_w   = (const float*)d_in[1];
    const float* gate_b   = (const float*)d_in[2];
    const float* expert_w = (const float*)d_in[3];
    const float* expert_b = (const float*)d_in[4];

    float* out        = (float*)d_out;
    float* out_pred   = out;                                       // B*C*H*W
    float* out_gate   = out_pred + (size_t)NBATCH * COUT * HW;     // B*K*H*W
    float* out_expert = out_gate + (size_t)NBATCH * NEXP * HW;     // B*K*C*H*W
    float* out_mask   = out_expert + (size_t)NBATCH * MTOT * HW;   // B*K*H*W

    const int npix_blocks = (NBATCH * HW) / 256;  // 512

    moe_gate_kernel<<<npix_blocks, 256, 0, stream>>>(
        x, gate_w, gate_b, out_gate, out_mask);

    dim3 cgrid(MTOT / MT, HH, NBATCH);  // (4, 128, 8)
    moe_expert_conv_wmma<<<cgrid, 256, 0, stream>>>(
        x, expert_w, expert_b, out_expert);

    moe_combine_kernel<<<npix_blocks, 256, 0, stream>>>(
        out_gate, out_expert, out_pred);
}